// V3Loss_41807211659561
// MI455X (gfx1250) — compile-verified
//
#include <hip/hip_runtime.h>

// V3 detection loss for MI455X (gfx1250), fully fused, fp32.
// - One kernel launch per scale; each thread owns one anchor.
// - GT boxes/labels staged in LDS; assignment recomputed per anchor
//   (64-bit class bitmask == multi-hot ts; ascending-m overwrite == last-write-wins tb).
// - cls/reg loads gated on fg (sparse): ~15-25 MB traffic instead of ~93 MB.
// - Intra-wave (32-lane) reduction done on the matrix pipe via v_wmma_f32_16x16x4_f32
//   (exact fp32: row-sum against an all-ones B matrix).

#define NUM_CLASSES 43
#define NMAX 50
#define IMG_SIZE 416.0f

typedef float v2f __attribute__((ext_vector_type(2)));
typedef float v8f __attribute__((ext_vector_type(8)));

__device__ __forceinline__ float bce_logits(float x, float t) {
    // stable BCE-with-logits, matches reference
    return fmaxf(x, 0.0f) - x * t + log1pf(expf(-fabsf(x)));
}

// Exact fp32 sum across the 32 lanes of a wave using the WMMA unit.
// A-matrix 16x4 layout: lane L<16 holds A[M=L][K=0..1], lane L>=16 holds
// A[M=L-16][K=2..3]. With B == ones(4x16), D[m][n] = v_m + v_{m+16}.
// C/D layout: lane L holds N=L%16, M = vgpr + 8*(L/16); so the 8-element
// local sum is sum(r_0..r_7) or sum(r_8..r_15); xor-16 completes the total.
__device__ __forceinline__ float wave_sum_wmma(float v) {
    v2f a;  a[0] = v;    a[1] = 0.0f;
    v2f bo; bo[0] = 1.0f; bo[1] = 1.0f;   // all-ones B: layout-invariant
    v8f c = {};
    v8f d = __builtin_amdgcn_wmma_f32_16x16x4_f32(
        /*neg_a=*/false, a, /*neg_b=*/false, bo,
        /*c_mod=*/(short)0, c, /*reuse_a=*/false, /*reuse_b=*/false);
    float s = d[0] + d[1] + d[2] + d[3] + d[4] + d[5] + d[6] + d[7];
    s += __shfl_xor(s, 16, 32);
    return s;
}

__global__ void __launch_bounds__(256)
v3_scale_loss(const float* __restrict__ cls,
              const float* __restrict__ reg,
              const float* __restrict__ obj,
              const float* __restrict__ bboxes,   // [B, 50, 4], normalized
              const int*   __restrict__ labels,   // [B, 50]
              float* __restrict__ acc,            // 4 floats: cls, box, obj, num_fg
              int W, int A, float stride_px)
{
    __shared__ float4 sbox[NMAX];
    __shared__ int    slab[NMAX];

    const int b   = blockIdx.y;
    const int tid = threadIdx.x;

    if (tid < NMAX) {
        float4 bb = reinterpret_cast<const float4*>(bboxes)[b * NMAX + tid];
        bb.x *= IMG_SIZE; bb.y *= IMG_SIZE; bb.z *= IMG_SIZE; bb.w *= IMG_SIZE;
        sbox[tid] = bb;
        slab[tid] = labels[b * NMAX + tid];
    }
    __syncthreads();

    const int a = blockIdx.x * blockDim.x + tid;

    float cls_s = 0.0f, box_s = 0.0f, obj_s = 0.0f, fg_s = 0.0f;

    if (a < A) {
        const int ax = a % W;
        const int ay = a / W;
        const float inv_s = 1.0f / stride_px;

        // ---- assignment (recomputed; replicates scan semantics) ----
        unsigned long long cmask = 0ull;
        float4 tb = make_float4(0.0f, 0.0f, 0.0f, 0.0f);
        #pragma unroll 5
        for (int m = 0; m < NMAX; ++m) {
            const int    lab = slab[m];
            const float4 bb  = sbox[m];
            const float cx = (bb.x + bb.z) * 0.5f;
            const float cy = (bb.y + bb.w) * 0.5f;
            int gx = (int)(cx * inv_s); gx = min(max(gx, 0), W - 1);
            int gy = (int)(cy * inv_s); gy = min(max(gy, 0), W - 1);
            const bool hit = (lab >= 0) &&
                             (abs(ax - gx) <= 1) && (abs(ay - gy) <= 1);
            if (hit) {
                cmask |= 1ull << min(max(lab, 0), NUM_CLASSES - 1);
                tb = bb;   // ascending m => last-write-wins, as in reference
            }
        }
        const bool  fg  = (cmask != 0ull);
        const float fgf = fg ? 1.0f : 0.0f;

        // ---- objectness BCE: dense over all anchors ----
        obj_s = bce_logits(obj[(size_t)b * A + a], fgf);
        fg_s  = fgf;

        if (fg) {
            // ---- box decode + IoU (fg only) ----
            const size_t rb = (size_t)b * 4 * A + (size_t)a;
            float rx = reg[rb];
            float ry = reg[rb + (size_t)A];
            float rw = reg[rb + (size_t)2 * A];
            float rh = reg[rb + (size_t)3 * A];
            const float pcx = ((float)ax + rx) * stride_px;
            const float pcy = ((float)ay + ry) * stride_px;
            rw = fminf(fmaxf(rw, -10.0f), 10.0f);
            rh = fminf(fmaxf(rh, -10.0f), 10.0f);
            const float pw = expf(rw) * stride_px;
            const float ph = expf(rh) * stride_px;
            const float px1 = pcx - 0.5f * pw, py1 = pcy - 0.5f * ph;
            const float px2 = pcx + 0.5f * pw, py2 = pcy + 0.5f * ph;

            const float ix1 = fmaxf(px1, tb.x), iy1 = fmaxf(py1, tb.y);
            const float ix2 = fminf(px2, tb.z), iy2 = fminf(py2, tb.w);
            const float inter = fmaxf(ix2 - ix1, 0.0f) * fmaxf(iy2 - iy1, 0.0f);
            const float pa = (px2 - px1) * (py2 - py1);
            const float ta = (tb.z - tb.x) * (tb.w - tb.y);
            const float iou = inter / (pa + ta - inter + 1e-7f);
            box_s = 1.0f - iou;

            // ---- focal classification loss (fg only; multi-hot target) ----
            const float* cp = cls + (size_t)b * NUM_CLASSES * A + (size_t)a;
            #pragma unroll 1
            for (int c = 0; c < NUM_CLASSES; ++c) {
                const int cn = (c + 1 < NUM_CLASSES) ? (c + 1) : c;
                __builtin_prefetch(cp + (size_t)cn * A, 0, 1);  // global_prefetch_b8
                const float x = cp[(size_t)c * A];
                const float t = (float)((cmask >> c) & 1ull);
                const float bce = bce_logits(x, t);
                const float pt  = expf(-bce);
                const float om  = 1.0f - pt;
                cls_s += 0.25f * om * om * bce;   // ALPHA=0.25, GAMMA=2
            }
        }
    }

    // ---- uniform control flow from here: WMMA reduction (EXEC all-1s) ----
    cls_s = wave_sum_wmma(cls_s);
    box_s = wave_sum_wmma(box_s);
    obj_s = wave_sum_wmma(obj_s);
    fg_s  = wave_sum_wmma(fg_s);

    if ((tid & 31) == 0) {
        atomicAdd(&acc[0], cls_s);
        atomicAdd(&acc[1], box_s);
        atomicAdd(&acc[2], obj_s);
        atomicAdd(&acc[3], fg_s);
    }
}

__global__ void init_acc(float* acc) {
    if (threadIdx.x < 12) acc[threadIdx.x] = 0.0f;
}

__global__ void finalize_loss(const float* __restrict__ acc, float* __restrict__ out) {
    if (threadIdx.x == 0 && blockIdx.x == 0) {
        float tc = 0.0f, tbx = 0.0f, to = 0.0f;
        #pragma unroll
        for (int s = 0; s < 3; ++s) {
            const float nf = fmaxf(acc[s * 4 + 3], 1.0f);
            tc  += acc[s * 4 + 0] / (nf * (float)NUM_CLASSES);
            tbx += acc[s * 4 + 1] / nf;
            to  += acc[s * 4 + 2] / nf;
        }
        // (LAMBDA_CLS*cls + LAMBDA_BOX*box + LAMBDA_OBJ*obj) / 3
        out[0] = (0.5f * tc + 7.5f * tbx + 1.0f * to) * (1.0f / 3.0f);
    }
}

extern "C" void kernel_launch(void* const* d_in, const int* in_sizes, int n_in,
                              void* d_out, int out_size, void* d_ws, size_t ws_size,
                              hipStream_t stream) {
    const float* cls[3] = { (const float*)d_in[0], (const float*)d_in[3], (const float*)d_in[6] };
    const float* reg[3] = { (const float*)d_in[1], (const float*)d_in[4], (const float*)d_in[7] };
    const float* obj[3] = { (const float*)d_in[2], (const float*)d_in[5], (const float*)d_in[8] };
    const float* bboxes = (const float*)d_in[9];
    const int*   labels = (const int*)d_in[10];

    const int Ws[3]       = { 52, 26, 13 };
    const float strides[3] = { 8.0f, 16.0f, 32.0f };

    // B derived from input sizes (setup: 128)
    const int B = in_sizes[0] / (NUM_CLASSES * Ws[0] * Ws[0]);

    float* acc = (float*)d_ws;              // 12 floats: {cls,box,obj,nfg} x 3 scales
    float* out = (float*)d_out;

    init_acc<<<1, 32, 0, stream>>>(acc);

    for (int s = 0; s < 3; ++s) {
        const int W = Ws[s];
        const int A = W * W;
        dim3 grid((A + 255) / 256, B);
        v3_scale_loss<<<grid, 256, 0, stream>>>(cls[s], reg[s], obj[s],
                                                bboxes, labels, acc + 4 * s,
                                                W, A, strides[s]);
    }

    finalize_loss<<<1, 1, 0, stream>>>(acc, out);
}